// ReadsOutLayer_4174708212123
// MI455X (gfx1250) — compile-verified
//
#include <hip/hip_runtime.h>
#include <hip/hip_bf16.h>

typedef float v2f __attribute__((ext_vector_type(2)));
typedef float v8f __attribute__((ext_vector_type(8)));

#define GRAPHS 128
#define DIM    256
#define BPG    16   // blocks per graph in the aggregation pass

// ---- monotone float <-> uint order mapping (for atomicMax on floats) ----
__device__ __forceinline__ unsigned fmap(float f) {
    unsigned b = __float_as_uint(f);
    return (b & 0x80000000u) ? ~b : (b | 0x80000000u);
}
__device__ __forceinline__ float funmap(unsigned u) {
    unsigned b = (u & 0x80000000u) ? (u & 0x7FFFFFFFu) : ~u;
    return __uint_as_float(b);
}

// ---- 0: init accumulators ----
__global__ void init_kernel(float* hp, unsigned* umax, int n) {
    int i = blockIdx.x * blockDim.x + threadIdx.x;
    if (i < n) {
        hp[i]   = 0.0f;
        umax[i] = fmap(-__builtin_inff());   // 0x007FFFFF
    }
}

// ---- 1: per-edge score via V_WMMA_F32_16X16X4_F32 ----
// One wave handles a 16-edge tile. A = 16x4 chunk of edge_feats (f32),
// B = 4x16 with W replicated into every column, so every column of the
// accumulated 16x16 D equals dot(feats_row, W). Lane 0 holds M=0..7 (N=0)
// in c[0..7]; lane 16 holds M=8..15.
__global__ void score_kernel(const float* __restrict__ feats,
                             const float* __restrict__ W,
                             const float* __restrict__ bias,
                             float* __restrict__ s, int E) {
    int gtid  = blockIdx.x * blockDim.x + threadIdx.x;
    int wave  = gtid >> 5;
    int lane  = threadIdx.x & 31;
    int tiles = (E + 15) >> 4;
    if (wave >= tiles) return;              // wave-uniform exit, EXEC stays full

    int row0 = wave << 4;
    int row  = row0 + (lane & 15);
    if (row >= E) row = E - 1;              // clamp (tail safety; 500000 % 16 == 0)
    const float* arow = feats + (long long)row * DIM;
    const int koff = (lane < 16) ? 0 : 2;   // A 16x4 f32 layout: lanes 16-31 carry K+2,K+3

    v8f c = {};
    #pragma unroll 4
    for (int k = 0; k < DIM; k += 4) {
        v2f a, b;
        a.x = arow[k + koff];
        a.y = arow[k + koff + 1];
        b.x = W[k + koff];                  // replicate W across all N columns
        b.y = W[k + koff + 1];
        c = __builtin_amdgcn_wmma_f32_16x16x4_f32(
                /*neg_a=*/false, a, /*neg_b=*/false, b,
                /*c_mod=*/(short)0, c, /*reuse_a=*/false, /*reuse_b=*/false);
    }

    float bv = bias[0];
    if (lane == 0 || lane == 16) {
        int mbase = row0 + ((lane == 16) ? 8 : 0);
        #pragma unroll
        for (int i = 0; i < 8; ++i) {
            int r = mbase + i;
            if (r < E) {
                float v = c[i] + bv;
                s[r] = (v > 0.0f) ? v : 0.01f * v;   // LeakyReLU(0.01)
            }
        }
    }
}

// ---- 2: segment boundaries from sorted ids ----
__global__ void bounds_kernel(const int* __restrict__ ids, int* __restrict__ offs,
                              int E, int G) {
    int e = blockIdx.x * blockDim.x + threadIdx.x;
    if (e >= E) return;
    int cur  = ids[e];
    int prev = (e == 0) ? -1 : ids[e - 1];
    for (int g = prev + 1; g <= cur; ++g) offs[g] = e;
    if (e == E - 1)
        for (int g = cur + 1; g <= G; ++g) offs[g] = E;
}

// ---- 3: per-graph softmax stats: max and 1/sum(exp) ----
__global__ void stats_kernel(const float* __restrict__ s, const int* __restrict__ offs,
                             float* __restrict__ smax, float* __restrict__ dinv) {
    __shared__ float red[256];
    int g = blockIdx.x;
    int start = offs[g], end = offs[g + 1];
    int t = threadIdx.x;

    float m = -__builtin_inff();
    for (int e = start + t; e < end; e += 256) m = fmaxf(m, s[e]);
    red[t] = m;
    __syncthreads();
    for (int o = 128; o > 0; o >>= 1) {
        if (t < o) red[t] = fmaxf(red[t], red[t + o]);
        __syncthreads();
    }
    m = red[0];
    __syncthreads();

    float sum = 0.0f;
    for (int e = start + t; e < end; e += 256) sum += expf(s[e] - m);
    red[t] = sum;
    __syncthreads();
    for (int o = 128; o > 0; o >>= 1) {
        if (t < o) red[t] += red[t + o];
        __syncthreads();
    }
    if (t == 0) {
        smax[g] = m;
        dinv[g] = (red[0] > 0.0f) ? 1.0f / red[0] : 0.0f;
    }
}

// ---- 4: heavy pass — weighted feature sum + feature max per graph ----
// Thread t owns dimension t; 256 threads read a 1KB row fully coalesced.
__global__ void agg_kernel(const float* __restrict__ feats, const float* __restrict__ s,
                           const int* __restrict__ offs, const float* __restrict__ smax,
                           const float* __restrict__ dinv,
                           float* __restrict__ hp, unsigned* __restrict__ umax) {
    int g   = blockIdx.x / BPG;
    int sub = blockIdx.x % BPG;
    int start = offs[g], end = offs[g + 1];
    if (end <= start) return;
    int t = threadIdx.x;

    float m  = smax[g];
    float di = dinv[g];
    float acc = 0.0f;
    float mx  = -__builtin_inff();
    for (int e = start + sub; e < end; e += BPG) {
        float w = expf(s[e] - m) * di;              // broadcast load of s[e]
        float f = feats[(long long)e * DIM + t];
        acc += w * f;
        mx  = fmaxf(mx, f);
    }
    atomicAdd(&hp[g * DIM + t], acc);
    atomicMax(&umax[g * DIM + t], fmap(mx));
}

// ---- 5: write [G, 2D] output ----
__global__ void fin_kernel(const float* __restrict__ hp, const unsigned* __restrict__ umax,
                           float* __restrict__ out, int n) {
    int idx = blockIdx.x * blockDim.x + threadIdx.x;   // g*256 + d
    if (idx >= n) return;
    int g = idx >> 8, d = idx & 255;
    out[g * (2 * DIM) + d]       = hp[idx];
    out[g * (2 * DIM) + DIM + d] = funmap(umax[idx]);
}

static inline size_t align256(size_t x) { return (x + 255) & ~(size_t)255; }

extern "C" void kernel_launch(void* const* d_in, const int* in_sizes, int n_in,
                              void* d_out, int out_size, void* d_ws, size_t ws_size,
                              hipStream_t stream) {
    const float* feats = (const float*)d_in[0];   // [E, 256]
    const float* W     = (const float*)d_in[1];   // [256, 1]
    const float* bias  = (const float*)d_in[2];   // [1]
    const int*   ids   = (const int*)d_in[3];     // [E], sorted
    const int E = in_sizes[3];
    const int G = GRAPHS;

    // workspace layout
    char* ws = (char*)d_ws;
    size_t off = 0;
    float*    s    = (float*)(ws + off);    off += align256((size_t)E * sizeof(float));
    int*      offs = (int*)(ws + off);      off += align256((size_t)(G + 1) * sizeof(int));
    float*    smax = (float*)(ws + off);    off += align256((size_t)G * sizeof(float));
    float*    dinv = (float*)(ws + off);    off += align256((size_t)G * sizeof(float));
    float*    hp   = (float*)(ws + off);    off += align256((size_t)G * DIM * sizeof(float));
    unsigned* umax = (unsigned*)(ws + off); off += align256((size_t)G * DIM * sizeof(unsigned));
    (void)ws_size; (void)n_in; (void)out_size;

    // 0: init accumulators
    {
        int n = G * DIM;
        init_kernel<<<(n + 255) / 256, 256, 0, stream>>>(hp, umax, n);
    }
    // 1: WMMA scores
    {
        int tiles = (E + 15) >> 4;
        long long threads = (long long)tiles * 32;
        int blocks = (int)((threads + 255) / 256);
        score_kernel<<<blocks, 256, 0, stream>>>(feats, W, bias, s, E);
    }
    // 2: segment boundaries
    bounds_kernel<<<(E + 255) / 256, 256, 0, stream>>>(ids, offs, E, G);
    // 3: softmax stats
    stats_kernel<<<G, 256, 0, stream>>>(s, offs, smax, dinv);
    // 4: aggregation (bandwidth-dominant pass)
    agg_kernel<<<G * BPG, 256, 0, stream>>>(feats, s, offs, smax, dinv, hp, umax);
    // 5: finalize output
    {
        int n = G * DIM;
        fin_kernel<<<(n + 255) / 256, 256, 0, stream>>>(hp, umax, (float*)d_out, n);
    }
}